// TinyRNN_59133109731488
// MI455X (gfx1250) — compile-verified
//
#include <hip/hip_runtime.h>

// GRU scan for MI455X (gfx1250, wave32). fp32 WMMA (V_WMMA_F32_16X16X4_F32)
// keeps exact fp32 math vs the reference.
//
// Phase A (if workspace large enough): bulk x-projection GEMM, 65536 16x69
// tiles, stored transposed ([gate][m]) for direct consumption by the scan.
// Phase B: recurrent scan, one wave per 16-batch tile (128 waves), 512 steps
// of 30 chained f32 WMMAs + gate math; W_hh B-fragments are hoisted into
// registers (loop-invariant) so the serial chain only waits on the 6 per-step
// h A-fragments. Fallback: fused variant computes the x-projection in-loop.

typedef __attribute__((ext_vector_type(2))) float v2f;
typedef __attribute__((ext_vector_type(4))) float v4f;
typedef __attribute__((ext_vector_type(8))) float v8f;

#define B_N 2048
#define T_N 512
#define I_N 58
#define H_N 23
#define G_N 69          // 3*H
#define GP_TILES 5      // ceil(69/16) -> N padded to 80
#define KI_STEPS 15     // ceil(58/4)  -> K padded to 60
#define KH_STEPS 6      // ceil(23/4)  -> K padded to 24
#define XG_TILE 1104    // 69*16 floats per (batch-tile, t) in workspace

__device__ __forceinline__ v8f wmma_f32(v2f a, v2f b, v8f c) {
  // D = A(16x4 f32) x B(4x16 f32) + C(16x16 f32); wave32, EXEC must be all 1s.
  return __builtin_amdgcn_wmma_f32_16x16x4_f32(false, a, false, b,
                                               (short)0, c, false, false);
}

__device__ __forceinline__ float sigmoid_f(float v) {
  return 1.0f / (1.0f + __expf(-v));
}

// ---------------------------------------------------------------------------
// Phase A: x_proj[b, t, g] = sum_i x[b,t,i] * W_ih[g,i]   (bias added later)
// One wave computes one 16(batch) x 69(gate) tile at a fixed t.
// Output layout per tile: ws[tile][c][m], c = gate index, m = local batch row
// (this is exactly the lane-transposed layout the scan's gate math wants).
// ---------------------------------------------------------------------------
__global__ __launch_bounds__(256) void xproj_wmma_kernel(
    const float* __restrict__ x, const float* __restrict__ W_ih,
    float* __restrict__ ws) {
  __shared__ float smem[4800 + 8 * 960];  // WihB frags + 8 per-wave X tiles
  const int tid = threadIdx.x;

  // Build W_ih^T (K=58->60, N=69->80) in WMMA B-fragment layout, zero padded.
  // Fragment (kk,nn): 64 floats; lane l holds B[k=kk*4+koff+{0,1}][n=l%16],
  // koff = (l>=16)?2:0.  (Assumed B layout mirrors the documented A layout.)
  for (int s = tid; s < KI_STEPS * GP_TILES * 64; s += 256) {
    int kk = s / (GP_TILES * 64);
    int rem = s % (GP_TILES * 64);
    int nn = rem / 64;
    int l = rem % 64;
    int ln = l >> 1, j = l & 1;
    int n = ln & 15, ko = (ln >= 16) ? 2 : 0;
    int g = nn * 16 + n;
    int i = kk * 4 + ko + j;
    smem[s] = (g < G_N && i < I_N) ? W_ih[g * I_N + i] : 0.0f;
  }
  __syncthreads();

  const int lane = tid & 31, wid = tid >> 5;
  const int mr = lane & 15;
  const int koff = (lane >= 16) ? 2 : 0;
  const int m0 = (lane >= 16) ? 8 : 0;
  float* lx = &smem[4800 + wid * 960];  // 16 x 60 padded X tile

  const int job = blockIdx.x * 8 + wid;  // 0 .. 65535
  const int bb = job >> 9;               // batch tile 0..127
  const int t = job & (T_N - 1);
  const int b0 = bb * 16;

  if (lane < 16) { lx[lane * 60 + 58] = 0.0f; lx[lane * 60 + 59] = 0.0f; }
  // 16*58 = 928 = 29 * 32 exactly; coalesced within rows.
#pragma unroll
  for (int it = 0; it < 29; ++it) {
    int idx = it * 32 + lane;
    int r = idx / I_N, c = idx - r * I_N;
    lx[r * 60 + c] = x[((size_t)(b0 + r) * T_N + t) * I_N + c];
  }
  asm volatile("s_wait_dscnt 0" ::: "memory");  // cross-lane LDS visibility

  float* wsp = ws + (size_t)job * XG_TILE;
#pragma unroll
  for (int nn = 0; nn < GP_TILES; ++nn) {
    v8f acc = {0.f, 0.f, 0.f, 0.f, 0.f, 0.f, 0.f, 0.f};
#pragma unroll
    for (int kk = 0; kk < KI_STEPS; ++kk) {
      v2f a = *(const v2f*)&lx[mr * 60 + kk * 4 + koff];
      v2f b = *(const v2f*)&smem[(kk * GP_TILES + nn) * 64 + lane * 2];
      acc = wmma_f32(a, b, acc);
    }
    int c = nn * 16 + mr;  // this lane's gate column
    if (c < G_N) {         // divergent store AFTER all WMMAs of this tile
      *(v4f*)&wsp[c * 16 + m0]     = __builtin_shufflevector(acc, acc, 0, 1, 2, 3);
      *(v4f*)&wsp[c * 16 + m0 + 4] = __builtin_shufflevector(acc, acc, 4, 5, 6, 7);
    }
  }
}

// ---------------------------------------------------------------------------
// Phase B: sequential GRU scan. One wave (32 threads) per 16-batch tile.
// FUSED=true computes the x-projection in-loop (no workspace needed).
// ---------------------------------------------------------------------------
template <bool FUSED>
__global__ __launch_bounds__(32) void gru_scan_wmma_kernel(
    const float* __restrict__ x, const float* __restrict__ W_ih,
    const float* __restrict__ W_hh, const float* __restrict__ b_ih,
    const float* __restrict__ b_hh, const float* __restrict__ pi_w,
    const float* __restrict__ pi_b, const float* __restrict__ xg_ws,
    float* __restrict__ out) {
  __shared__ float smem[10816];
  constexpr int OFF_WIHB = 0;      // 15*5*64 = 4800
  constexpr int OFF_WHHB = 4800;   // 6*5*64  = 1920
  constexpr int OFF_BIH = 6720;    // 69
  constexpr int OFF_BHH = 6789;    // 69
  constexpr int OFF_PIW = 6858;    // 23
  constexpr int OFF_PIB = 6881;    // 23
  constexpr int OFF_LX = 6912;     // 16x60 padded X tile (fused only)
  constexpr int OFF_LH = 7872;     // 16x24 padded h (row-major, col 23 == 0)
  constexpr int OFF_XG = 8256;     // 80x16 xg transposed (fused only)
  constexpr int OFF_HG = 9536;     // 80x16 hg transposed

  const int lane = threadIdx.x & 31;
  const int mr = lane & 15;
  const int koff = (lane >= 16) ? 2 : 0;
  const int m0 = (lane >= 16) ? 8 : 0;
  const int b0 = blockIdx.x * 16;

  // ---- one-time setup: weights into WMMA B-fragment layout, biases, h=0 ----
  if constexpr (FUSED) {
    for (int s = lane; s < KI_STEPS * GP_TILES * 64; s += 32) {
      int kk = s / (GP_TILES * 64);
      int rem = s % (GP_TILES * 64);
      int nn = rem / 64;
      int l = rem % 64;
      int ln = l >> 1, j = l & 1;
      int n = ln & 15, ko = (ln >= 16) ? 2 : 0;
      int g = nn * 16 + n;
      int i = kk * 4 + ko + j;
      smem[OFF_WIHB + s] = (g < G_N && i < I_N) ? W_ih[g * I_N + i] : 0.0f;
    }
    if (lane < 16) { smem[OFF_LX + lane * 60 + 58] = 0.0f; smem[OFF_LX + lane * 60 + 59] = 0.0f; }
  }
  for (int s = lane; s < KH_STEPS * GP_TILES * 64; s += 32) {
    int kk = s / (GP_TILES * 64);
    int rem = s % (GP_TILES * 64);
    int nn = rem / 64;
    int l = rem % 64;
    int ln = l >> 1, j = l & 1;
    int n = ln & 15, ko = (ln >= 16) ? 2 : 0;
    int g = nn * 16 + n;
    int k = kk * 4 + ko + j;
    smem[OFF_WHHB + s] = (g < G_N && k < H_N) ? W_hh[g * H_N + k] : 0.0f;
  }
  for (int s = lane; s < G_N; s += 32) {
    smem[OFF_BIH + s] = b_ih[s];
    smem[OFF_BHH + s] = b_hh[s];
  }
  for (int s = lane; s < H_N; s += 32) {
    smem[OFF_PIW + s] = pi_w[s];
    smem[OFF_PIB + s] = pi_b[s];
  }
  for (int s = lane; s < 16 * 24; s += 32) smem[OFF_LH + s] = 0.0f;  // h0 = 0
  __syncthreads();

  // Hoist loop-invariant W_hh B-fragments into registers (30 x v2f = 60 VGPR).
  // Removes 30 LDS loads (and their dscnt waits) from every serial timestep.
  v2f whh[GP_TILES][KH_STEPS];
#pragma unroll
  for (int nn = 0; nn < GP_TILES; ++nn)
#pragma unroll
    for (int kk = 0; kk < KH_STEPS; ++kk)
      whh[nn][kk] =
          *(const v2f*)&smem[OFF_WHHB + (kk * GP_TILES + nn) * 64 + lane * 2];

  for (int t = 0; t < T_N; ++t) {
    if constexpr (FUSED) {
      // Load this step's 16x58 X tile into LDS (zero-padded to 60 cols).
#pragma unroll
      for (int it = 0; it < 29; ++it) {
        int idx = it * 32 + lane;
        int r = idx / I_N, c = idx - r * I_N;
        smem[OFF_LX + r * 60 + c] = x[((size_t)(b0 + r) * T_N + t) * I_N + c];
      }
      if (t + 1 < T_N && lane < 16) {  // prefetch next step's rows
        __builtin_prefetch(&x[((size_t)(b0 + lane) * T_N + (t + 1)) * I_N], 0, 1);
        __builtin_prefetch(&x[((size_t)(b0 + lane) * T_N + (t + 1)) * I_N + 32], 0, 1);
      }
    } else {
      if (t + 1 < T_N) {  // prefetch next step's xg tile (4416 B, 140 B/lane)
        __builtin_prefetch(
            xg_ws + ((size_t)blockIdx.x * T_N + (t + 1)) * XG_TILE + lane * 35, 0, 1);
      }
    }
    // Make x-tile stores and previous step's h updates visible wave-wide.
    asm volatile("s_wait_dscnt 0" ::: "memory");

    if constexpr (FUSED) {
      // xg = X @ W_ih^T : 5 N-tiles x 15 K-steps of f32 WMMA.
#pragma unroll
      for (int nn = 0; nn < GP_TILES; ++nn) {
        v8f acc = {0.f, 0.f, 0.f, 0.f, 0.f, 0.f, 0.f, 0.f};
#pragma unroll
        for (int kk = 0; kk < KI_STEPS; ++kk) {
          v2f a = *(const v2f*)&smem[OFF_LX + mr * 60 + kk * 4 + koff];
          v2f b = *(const v2f*)&smem[OFF_WIHB + (kk * GP_TILES + nn) * 64 + lane * 2];
          acc = wmma_f32(a, b, acc);
        }
        int c = nn * 16 + mr;
        *(v4f*)&smem[OFF_XG + c * 16 + m0]     = __builtin_shufflevector(acc, acc, 0, 1, 2, 3);
        *(v4f*)&smem[OFF_XG + c * 16 + m0 + 4] = __builtin_shufflevector(acc, acc, 4, 5, 6, 7);
      }
    }

    // hg = h @ W_hh^T : 5 N-tiles x 6 K-steps of f32 WMMA.
    // A-fragments of h: only 6 LDS loads on the serial chain; B from registers.
    v2f ha[KH_STEPS];
#pragma unroll
    for (int kk = 0; kk < KH_STEPS; ++kk)
      ha[kk] = *(const v2f*)&smem[OFF_LH + mr * 24 + kk * 4 + koff];
#pragma unroll
    for (int nn = 0; nn < GP_TILES; ++nn) {
      v8f acc = {0.f, 0.f, 0.f, 0.f, 0.f, 0.f, 0.f, 0.f};
#pragma unroll
      for (int kk = 0; kk < KH_STEPS; ++kk) acc = wmma_f32(ha[kk], whh[nn][kk], acc);
      int c = nn * 16 + mr;
      *(v4f*)&smem[OFF_HG + c * 16 + m0]     = __builtin_shufflevector(acc, acc, 0, 1, 2, 3);
      *(v4f*)&smem[OFF_HG + c * 16 + m0 + 4] = __builtin_shufflevector(acc, acc, 4, 5, 6, 7);
    }
    asm volatile("s_wait_dscnt 0" ::: "memory");

    // Gate math: 16 rows x 23 units = 368 elements over 32 lanes.
    const float* xgp =
        FUSED ? nullptr : (xg_ws + ((size_t)blockIdx.x * T_N + t) * XG_TILE);
    for (int it = 0; it < 12; ++it) {
      int idx = it * 32 + lane;
      if (idx < 16 * H_N) {
        int m = idx / H_N;
        int u = idx - m * H_N;
        float xr, xz, xn;
        if constexpr (FUSED) {
          xr = smem[OFF_XG + u * 16 + m];
          xz = smem[OFF_XG + (H_N + u) * 16 + m];
          xn = smem[OFF_XG + (2 * H_N + u) * 16 + m];
        } else {
          xr = xgp[u * 16 + m];
          xz = xgp[(H_N + u) * 16 + m];
          xn = xgp[(2 * H_N + u) * 16 + m];
        }
        xr += smem[OFF_BIH + u];
        xz += smem[OFF_BIH + H_N + u];
        xn += smem[OFF_BIH + 2 * H_N + u];
        float hr = smem[OFF_HG + u * 16 + m] + smem[OFF_BHH + u];
        float hz = smem[OFF_HG + (H_N + u) * 16 + m] + smem[OFF_BHH + H_N + u];
        float hn = smem[OFF_HG + (2 * H_N + u) * 16 + m] + smem[OFF_BHH + 2 * H_N + u];
        float rg = sigmoid_f(xr + hr);
        float zg = sigmoid_f(xz + hz);
        float ng = tanhf(xn + rg * hn);
        float hprev = smem[OFF_LH + m * 24 + u];
        float hnew = (1.0f - zg) * ng + zg * hprev;
        smem[OFF_LH + m * 24 + u] = hnew;
        out[((size_t)(b0 + m) * T_N + t) * H_N + u] =
            smem[OFF_PIW + u] * hnew + smem[OFF_PIB + u];
      }
    }
  }

  // Final hidden state hT -> out[B*T*H + b*H + u].
  asm volatile("s_wait_dscnt 0" ::: "memory");
  const size_t ht_off = (size_t)B_N * T_N * H_N;
  for (int it = 0; it < 12; ++it) {
    int idx = it * 32 + lane;
    if (idx < 16 * H_N) {
      int m = idx / H_N;
      int u = idx - m * H_N;
      out[ht_off + (size_t)(b0 + m) * H_N + u] = smem[OFF_LH + m * 24 + u];
    }
  }
}

extern "C" void kernel_launch(void* const* d_in, const int* in_sizes, int n_in,
                              void* d_out, int out_size, void* d_ws, size_t ws_size,
                              hipStream_t stream) {
  (void)in_sizes; (void)n_in; (void)out_size;
  const float* x    = (const float*)d_in[0];
  const float* W_ih = (const float*)d_in[1];
  const float* W_hh = (const float*)d_in[2];
  const float* b_ih = (const float*)d_in[3];
  const float* b_hh = (const float*)d_in[4];
  const float* pi_w = (const float*)d_in[5];
  const float* pi_b = (const float*)d_in[6];
  float* out = (float*)d_out;
  float* ws  = (float*)d_ws;

  const size_t need = (size_t)(B_N / 16) * T_N * XG_TILE * sizeof(float);  // ~289 MB
  if (ws_size >= need) {
    // Phase A: bulk x-projection (65536 tiles, 8 waves/WG).
    xproj_wmma_kernel<<<(B_N / 16) * T_N / 8, 256, 0, stream>>>(x, W_ih, ws);
    // Phase B: scan with precomputed x-projection (shorter serial chain).
    gru_scan_wmma_kernel<false><<<B_N / 16, 32, 0, stream>>>(
        x, W_ih, W_hh, b_ih, b_hh, pi_w, pi_b, ws, out);
  } else {
    gru_scan_wmma_kernel<true><<<B_N / 16, 32, 0, stream>>>(
        x, W_ih, W_hh, b_ih, b_hh, pi_w, pi_b, nullptr, out);
  }
}